// Model_60559038874006
// MI455X (gfx1250) — compile-verified
//
#include <hip/hip_runtime.h>

// ---------------------------------------------------------------------------
// Fused 2-layer LSTM (T=2048, B=4096, I=H=4) for gfx1250 using
// V_WMMA_F32_16X16X4_F32. One wave32 owns 16 batch columns; the per-step gate
// GEMM [16x4]@[4x16] maps 1:1 onto the f32 WMMA shape. Weight rows are
// permuted so gates land lane-local => no cross-lane traffic per step.
// This version software-pipelines: body(t) = { layer1(t) , layer0(t+1) },
// two independent WMMA+TRANS chains the scheduler can interleave, and the
// x-load for t+1 is issued at the top of body(t), just before layer0(t+1) uses it.
// ---------------------------------------------------------------------------

typedef __attribute__((ext_vector_type(2))) float v2f;
typedef __attribute__((ext_vector_type(8))) float v8f;

#define T_LEN 2048
#define B_LEN 4096

__device__ __forceinline__ float fast_tanh(float x) {
#if __has_builtin(__builtin_amdgcn_tanhf)
  return __builtin_amdgcn_tanhf(x);               // v_tanh_f32 (TRANS pipe)
#elif __has_builtin(__builtin_amdgcn_tanh_f32)
  return __builtin_amdgcn_tanh_f32(x);
#else
  float e = __expf(-2.0f * x);
  float r = __builtin_amdgcn_rcpf(1.0f + e);
  return fmaf(-2.0f * e, r, 1.0f);                // (1-e)/(1+e)
#endif
}

__device__ __forceinline__ float fast_sigmoid(float x) {
  return fmaf(0.5f, fast_tanh(0.5f * x), 0.5f);
}

// One LSTM cell step for this lane's 16-batch tile.
// in_b : B-operand (input activations, lane-local K-half)
// h, c : recurrent state, lane-local hidden units {2*hi, 2*hi+1}
__device__ __forceinline__ void lstm_cell(v2f a_ih, v2f a_hh, v8f cb,
                                          v2f in_b, v2f& h, v2f& c) {
  v8f g = cb;
  g = __builtin_amdgcn_wmma_f32_16x16x4_f32(false, a_ih, false, in_b,
                                            (short)0, g, false, false);
  g = __builtin_amdgcn_wmma_f32_16x16x4_f32(false, a_hh, false, h,
                                            (short)0, g, false, false);
  // lane-local gates (permuted rows): g[0..1]=i, g[2..3]=f, g[4..5]=g, g[6..7]=o
#pragma unroll
  for (int u = 0; u < 2; ++u) {
    const float ig = fast_sigmoid(g[u]);
    const float fg = fast_sigmoid(g[2 + u]);
    const float gg = fast_tanh(g[4 + u]);
    const float og = fast_sigmoid(g[6 + u]);
    const float cp = u ? c.y : c.x;
    const float cc = fmaf(fg, cp, ig * gg);
    const float hh = og * fast_tanh(cc);
    if (u) { c.y = cc; h.y = hh; } else { c.x = cc; h.x = hh; }
  }
}

__global__ __launch_bounds__(32) void lstm2_wmma_f32_kernel(
    const float* __restrict__ x,
    const float* __restrict__ Wih0, const float* __restrict__ Whh0,
    const float* __restrict__ bih0, const float* __restrict__ bhh0,
    const float* __restrict__ Wih1, const float* __restrict__ Whh1,
    const float* __restrict__ bih1, const float* __restrict__ bhh1,
    float* __restrict__ out) {
  const int lane = threadIdx.x & 31;
  const int n    = lane & 15;      // batch column within tile / A row index
  const int hi   = lane >> 4;      // K-half for A/B, M-half for C/D
  const int kb   = 2 * hi;         // first K (and hidden-unit) index this lane owns
  const int bt   = blockIdx.x * 16;

  // permuted row M=n -> original PyTorch gate row
  // perm = [i0,i1,f0,f1,g0,g1,o0,o1, i2,i3,f2,f3,g2,g3,o2,o3]
  const int orow = ((n & 7) >> 1) * 4 + (n & 1) + 2 * (n >> 3);

  // ---- loop-invariant A operands: permuted weight rows, K-half per lane ----
  v2f a_ih0, a_hh0, a_ih1, a_hh1;
  a_ih0.x = Wih0[orow * 4 + kb];  a_ih0.y = Wih0[orow * 4 + kb + 1];
  a_hh0.x = Whh0[orow * 4 + kb];  a_hh0.y = Whh0[orow * 4 + kb + 1];
  a_ih1.x = Wih1[orow * 4 + kb];  a_ih1.y = Wih1[orow * 4 + kb + 1];
  a_hh1.x = Whh1[orow * 4 + kb];  a_hh1.y = Whh1[orow * 4 + kb + 1];

  // ---- loop-invariant bias as C operand (C[m,*] = bias[perm(m)]) ----
  v8f cb0, cb1;
#pragma unroll
  for (int v = 0; v < 8; ++v) {
    const int r = (v >> 1) * 4 + (v & 1) + 2 * hi;  // original bias row
    cb0[v] = bih0[r] + bhh0[r];
    cb1[v] = bih1[r] + bhh1[r];
  }

  // ---- recurrent state, lane-local ----
  v2f h0; h0.x = 0.0f; h0.y = 0.0f;
  v2f c0 = h0, h1 = h0, c1 = h0;

  const size_t stride = (size_t)B_LEN * 4;                 // floats per timestep
  const float* xp = x   + (size_t)(bt + n) * 4 + kb;       // B-operand slot
  float*       op = out + (size_t)(bt + n) * 4 + kb;

  // -------- prologue: layer0 step t=0 --------
  v2f bx = __builtin_nontemporal_load((const v2f*)xp);
  lstm_cell(a_ih0, a_hh0, cb0, bx, h0, c0);

  // -------- skewed steady state: body(t) = { layer1(t), layer0(t+1) } ------
  for (int t = 0; t < T_LEN - 1; ++t) {
    // x(t+1): loaded here and used right away by layer0 below
    v2f bxn = __builtin_nontemporal_load((const v2f*)(xp + stride));
    if (t + 9 < T_LEN) __builtin_prefetch(xp + 9 * stride, 0, 1);

    const v2f h0_cur = h0;  // layer0 output at step t (layer1's input)

    // layer0 step t+1  — independent of layer1 step t below
    lstm_cell(a_ih0, a_hh0, cb0, bxn, h0, c0);
    // layer1 step t
    lstm_cell(a_ih1, a_hh1, cb1, h0_cur, h1, c1);

    __builtin_nontemporal_store(h1, (v2f*)op);  // out[t, bt+n, kb:kb+2]

    xp += stride;
    op += stride;
  }

  // -------- epilogue: layer1 step T-1 --------
  lstm_cell(a_ih1, a_hh1, cb1, h0, h1, c1);
  __builtin_nontemporal_store(h1, (v2f*)op);
}

extern "C" void kernel_launch(void* const* d_in, const int* in_sizes, int n_in,
                              void* d_out, int out_size, void* d_ws,
                              size_t ws_size, hipStream_t stream) {
  (void)in_sizes; (void)n_in; (void)out_size; (void)d_ws; (void)ws_size;
  const float* x    = (const float*)d_in[0];
  const float* Wih0 = (const float*)d_in[1];
  const float* Whh0 = (const float*)d_in[2];
  const float* bih0 = (const float*)d_in[3];
  const float* bhh0 = (const float*)d_in[4];
  const float* Wih1 = (const float*)d_in[5];
  const float* Whh1 = (const float*)d_in[6];
  const float* bih1 = (const float*)d_in[7];
  const float* bhh1 = (const float*)d_in[8];
  float* out = (float*)d_out;

  // 4096 batch / 16 per wave = 256 single-wave (wave32) workgroups.
  lstm2_wmma_f32_kernel<<<dim3(B_LEN / 16), dim3(32), 0, stream>>>(
      x, Wih0, Whh0, bih0, bhh0, Wih1, Whh1, bih1, bhh1, out);
}